// FasterRcnn_72662256714231
// MI455X (gfx1250) — compile-verified
//
#include <hip/hip_runtime.h>
#include <hip/hip_bf16.h>
#include <math.h>

#define N_ROI   2000
#define N_CLS   21
#define N_FG    20
#define NMS_T   0.3f
#define SCORE_T 0.05f
#define KWORDS  ((N_ROI + 31) / 32)

typedef __attribute__((ext_vector_type(2))) float v2f;
typedef __attribute__((ext_vector_type(8))) float v8f;

// ---------------------------------------------------------------------------
// Kernel A: per-ROI box decoding + WMMA-based softmax (one wave handles 16 ROIs)
// ---------------------------------------------------------------------------
__global__ __launch_bounds__(32)
void decode_softmax_kernel(const float* __restrict__ rois,
                           const float* __restrict__ locs,
                           const float* __restrict__ scores,
                           const float* __restrict__ scale,
                           float* __restrict__ boxes_c,   // [20][N][4] class-major
                           float* __restrict__ probs_c)   // [20][N]
{
    __shared__ float sexp[16][24];   // exp numerators, padded K=24
    __shared__ float ssum[16];
    const int lane = threadIdx.x;
    const int base = blockIdx.x * 16;          // 125 blocks * 16 = 2000 exactly
    const float inv_scale = 1.0f / scale[0];
    const float maxy = 600.0f * inv_scale;
    const float maxx = 800.0f * inv_scale;

    if (lane < 16) {
        const int n = base + lane;
        // softmax numerators (max-subtracted)
        float m = -INFINITY;
        float sc[N_CLS];
        for (int c = 0; c < N_CLS; ++c) { sc[c] = scores[n * N_CLS + c]; m = fmaxf(m, sc[c]); }
        for (int c = 0; c < N_CLS; ++c) sexp[lane][c] = expf(sc[c] - m);
        for (int c = N_CLS; c < 24; ++c) sexp[lane][c] = 0.0f;

        // decode + clip boxes for foreground classes 1..20
        const float y1 = rois[n * 4 + 0], x1 = rois[n * 4 + 1];
        const float y2 = rois[n * 4 + 2], x2 = rois[n * 4 + 3];
        const float h = y2 - y1, w = x2 - x1;
        const float cy = y1 + 0.5f * h, cx = x1 + 0.5f * w;
        for (int c = 1; c < N_CLS; ++c) {
            const float dy = locs[n * (N_CLS * 4) + c * 4 + 0] * 0.1f;
            const float dx = locs[n * (N_CLS * 4) + c * 4 + 1] * 0.1f;
            const float dh = locs[n * (N_CLS * 4) + c * 4 + 2] * 0.2f;
            const float dw = locs[n * (N_CLS * 4) + c * 4 + 3] * 0.2f;
            const float ncy = dy * h + cy, ncx = dx * w + cx;
            const float nh = expf(dh) * h, nw = expf(dw) * w;
            const float by1 = fminf(fmaxf(ncy - 0.5f * nh, 0.0f), maxy);
            const float bx1 = fminf(fmaxf(ncx - 0.5f * nw, 0.0f), maxx);
            const float by2 = fminf(fmaxf(ncy + 0.5f * nh, 0.0f), maxy);
            const float bx2 = fminf(fmaxf(ncx + 0.5f * nw, 0.0f), maxx);
            const size_t o = ((size_t)(c - 1) * N_ROI + n) * 4;
            boxes_c[o + 0] = by1; boxes_c[o + 1] = bx1;
            boxes_c[o + 2] = by2; boxes_c[o + 3] = bx2;
        }
    }
    __syncthreads();

    // WMMA row-sum: D = A x Ones + D ; A is 16x4 f32 chunk of the exp matrix.
    // A layout (16x4 f32): lanes 0-15 -> K0/K1 in the 2 VGPRs, lanes 16-31 -> K2/K3.
    const int am = lane & 15;
    const int kb = (lane >> 4) * 2;
    v8f acc = {0.f, 0.f, 0.f, 0.f, 0.f, 0.f, 0.f, 0.f};
    v2f bones; bones.x = 1.0f; bones.y = 1.0f;
    #pragma unroll
    for (int ch = 0; ch < 6; ++ch) {          // K = 6*4 = 24 (padded 21)
        v2f a;
        a.x = sexp[am][ch * 4 + kb];
        a.y = sexp[am][ch * 4 + kb + 1];
        acc = __builtin_amdgcn_wmma_f32_16x16x4_f32(
            /*neg_a=*/false, a, /*neg_b=*/false, bones,
            /*c_mod=*/(short)0, acc, /*reuse_a=*/false, /*reuse_b=*/false);
    }
    // C/D layout: lane 0 holds (M=0..7, N=0) in VGPR 0..7; lane 16 holds M=8..15.
    if (lane == 0)  { for (int r = 0; r < 8; ++r) ssum[r]     = acc[r]; }
    if (lane == 16) { for (int r = 0; r < 8; ++r) ssum[8 + r] = acc[r]; }
    __syncthreads();

    if (lane < 16) {
        const int n = base + lane;
        const float inv = 1.0f / ssum[lane];
        for (int c = 1; c < N_CLS; ++c)
            probs_c[(size_t)(c - 1) * N_ROI + n] = sexp[lane][c] * inv;
    }
}

// ---------------------------------------------------------------------------
// Kernel B: per-class stable descending rank (== stable argsort(-key))
// ---------------------------------------------------------------------------
__global__ __launch_bounds__(256)
void rank_kernel(const float* __restrict__ probs_c, int* __restrict__ order)
{
    __shared__ float key[N_ROI];
    const int c = blockIdx.x;
    for (int i = threadIdx.x; i < N_ROI; i += blockDim.x) {
        const float p = probs_c[(size_t)c * N_ROI + i];
        key[i] = (p > SCORE_T) ? p : -INFINITY;
    }
    __syncthreads();
    for (int i = threadIdx.x; i < N_ROI; i += blockDim.x) {
        const float ki = key[i];
        int r = 0;
        for (int j = 0; j < N_ROI; ++j) {
            const float kj = key[j];
            r += (int)((kj > ki) || ((kj == ki) && (j < i)));
        }
        order[(size_t)c * N_ROI + r] = i;
    }
}

// ---------------------------------------------------------------------------
// Kernel C: per-class greedy NMS (bitmask in LDS) + output scatter
// ---------------------------------------------------------------------------
__global__ __launch_bounds__(256)
void nms_kernel(const float* __restrict__ boxes_c,
                const float* __restrict__ probs_c,
                const int* __restrict__ order,
                float* __restrict__ out5,      // [20][N][5]
                float* __restrict__ keep_out,  // [20][N]
                float* __restrict__ label_out) // [20][N]
{
    __shared__ float4   sbox[N_ROI];     // sorted boxes (32 KB)
    __shared__ float    sprob[N_ROI];    // sorted probs
    __shared__ float    sarea[N_ROI];
    __shared__ unsigned keepw[KWORDS];
    const int c = blockIdx.x;
    const int tid = threadIdx.x;

    // Gather sorted boxes into LDS via async global->LDS B128 (TDM async path).
    for (int i = tid; i < N_ROI; i += blockDim.x) {
        const int o = order[(size_t)c * N_ROI + i];
        const float* src = boxes_c + ((size_t)c * N_ROI + o) * 4;
        const unsigned lds_addr = (unsigned)(unsigned long long)(uintptr_t)&sbox[i];
        const unsigned long long ga = (unsigned long long)(uintptr_t)src;
        asm volatile("global_load_async_to_lds_b128 %0, %1, off"
                     :: "v"(lds_addr), "v"(ga) : "memory");
        sprob[i] = probs_c[(size_t)c * N_ROI + o];
    }
    asm volatile("s_wait_asynccnt 0" ::: "memory");
    __syncthreads();

    for (int w = tid; w < KWORDS; w += blockDim.x) keepw[w] = 0u;
    __syncthreads();
    for (int i = tid; i < N_ROI; i += blockDim.x) {
        const float4 b = sbox[i];
        sarea[i] = (b.z - b.x) * (b.w - b.y);
        if (sprob[i] > SCORE_T) atomicOr(&keepw[i >> 5], 1u << (i & 31));
    }
    __syncthreads();

    // Greedy sweep: row i suppresses columns j > i. Keep-bit i is only written
    // by iterations < i (fenced by barrier), so `ki` is workgroup-uniform.
    for (int i = 0; i < N_ROI; ++i) {
        const bool ki = (keepw[i >> 5] >> (i & 31)) & 1u;
        if (ki) {
            const float4 bi = sbox[i];
            const float ai = sarea[i];
            for (int j = i + 1 + tid; j < N_ROI; j += blockDim.x) {
                const float4 bj = sbox[j];
                const float ty = fmaxf(bi.x, bj.x);
                const float tx = fmaxf(bi.y, bj.y);
                const float by = fminf(bi.z, bj.z);
                const float bx = fminf(bi.w, bj.w);
                const float inter = fmaxf(by - ty, 0.0f) * fmaxf(bx - tx, 0.0f);
                const float iou = inter / (ai + sarea[j] - inter + 1e-12f);
                if (iou > NMS_T) atomicAnd(&keepw[j >> 5], ~(1u << (j & 31)));
            }
            __syncthreads();   // uniform: ki identical across the workgroup
        }
    }

    // Scatter results back to original ROI order (order[] is a permutation,
    // so every output slot is written -- d_out is poisoned before timing).
    for (int i = tid; i < N_ROI; i += blockDim.x) {
        const int o = order[(size_t)c * N_ROI + i];
        const bool kept = (keepw[i >> 5] >> (i & 31)) & 1u;
        const float4 b = sbox[i];
        const float p = sprob[i];
        const size_t ob = ((size_t)c * N_ROI + o) * 5;
        out5[ob + 0] = kept ? b.x : 0.0f;
        out5[ob + 1] = kept ? b.y : 0.0f;
        out5[ob + 2] = kept ? b.z : 0.0f;
        out5[ob + 3] = kept ? b.w : 0.0f;
        out5[ob + 4] = kept ? p   : 0.0f;
        keep_out[(size_t)c * N_ROI + o]  = kept ? 1.0f : 0.0f;
        label_out[(size_t)c * N_ROI + o] = (float)c;
    }
}

// ---------------------------------------------------------------------------
extern "C" void kernel_launch(void* const* d_in, const int* in_sizes, int n_in,
                              void* d_out, int out_size, void* d_ws, size_t ws_size,
                              hipStream_t stream) {
    const float* rois   = (const float*)d_in[0];   // [2000, 4]
    const float* locs   = (const float*)d_in[1];   // [2000, 84]
    const float* scores = (const float*)d_in[2];   // [2000, 21]
    const float* scale  = (const float*)d_in[3];   // [1]

    float* ws      = (float*)d_ws;
    float* boxes_c = ws;                    // 20*2000*4 = 160000 floats
    float* probs_c = ws + 160000;           // 20*2000   =  40000 floats
    int*   order   = (int*)(ws + 200000);   // 20*2000   =  40000 ints

    float* out5  = (float*)d_out;           // 20*2000*5 = 200000
    float* keepf = out5 + 200000;           // 20*2000   =  40000
    float* labf  = keepf + 40000;           // 20*2000   =  40000

    decode_softmax_kernel<<<N_ROI / 16, 32, 0, stream>>>(rois, locs, scores, scale,
                                                         boxes_c, probs_c);
    rank_kernel<<<N_FG, 256, 0, stream>>>(probs_c, order);
    nms_kernel<<<N_FG, 256, 0, stream>>>(boxes_c, probs_c, order, out5, keepf, labf);
}